// RGCNDirectional_12300786335885
// MI455X (gfx1250) — compile-verified
//
#include <hip/hip_runtime.h>

typedef __attribute__((ext_vector_type(2))) float v2f;
typedef __attribute__((ext_vector_type(8))) float v8f;
typedef __attribute__((ext_vector_type(4))) int   v4i;

typedef __attribute__((address_space(1))) v4i* gptr_b128;  // global int4*
typedef __attribute__((address_space(3))) v4i* lptr_b128;  // LDS int4*

#define EPS 1e-12f
#define KT  128      // K-chunk of A staged in LDS
#define LDA 132      // padded LDS row stride (dwords): 4*m advance -> conflict-free b64

#if __has_builtin(__builtin_amdgcn_global_load_async_to_lds_b128) && \
    __has_builtin(__builtin_amdgcn_s_wait_asynccnt)
#define HAS_ASYNC_LDS 1
#else
#define HAS_ASYNC_LDS 0
#endif

// ---------------------------------------------------------------------------
// Kernel 1: HTf = fragment-packed H^T, where H[r] = X @ W[r]^T  (H^T = W @ X^T)
// Storage: dword index = r*524288 + (k>>2)*512 + c*4 + (k&3)   (c = out col, k = node)
// ---------------------------------------------------------------------------
__global__ __launch_bounds__(256) void rgcn_hproj(
    const float* __restrict__ X,    // [4096][128]
    const float* __restrict__ W,    // [8][128][128]
    float* __restrict__ HTf)        // 4M floats (16 MB) in d_ws
{
  const int lane = threadIdx.x & 31;
  const int wv   = threadIdx.x >> 5;     // 0..7  -> o-tile
  const int r    = blockIdx.x >> 8;      // 0..7
  const int nt   = blockIdx.x & 255;     // node tile
  const int k0   = nt * 16;
  const int o0   = wv * 16;

  const int mrow = lane & 15;
  const int half = lane >> 4;            // selects K pair {0,1} vs {2,3}

  const float* wp = W + (size_t)r * 16384 + (size_t)(o0 + mrow) * 128 + 2 * half;
  const float* xp = X + (size_t)(k0 + mrow) * 128 + 2 * half;

  v8f acc = {};
  #pragma unroll
  for (int d0 = 0; d0 < 128; d0 += 4) {
    v2f a = *(const v2f*)(wp + d0);      // A = W tile (16 x 4)
    v2f b = *(const v2f*)(xp + d0);      // B = X^T tile (4 x 16)
    acc = __builtin_amdgcn_wmma_f32_16x16x4_f32(false, a, false, b,
                                                (short)0, acc, false, false);
  }

  // D[v] = H^T[o0 + v + 8*half][k0 + mrow]
  const int k = k0 + mrow;
  float* hp = HTf + (size_t)r * 524288 + (size_t)(k >> 2) * 512 + (k & 3);
  #pragma unroll
  for (int v = 0; v < 8; ++v) {
    const int c = o0 + v + 8 * half;
    hp[c * 4] = acc[v];
  }
}

// ---------------------------------------------------------------------------
// Kernel 2: out[m,:] = sum_r ( A[r][m,:] @ H[r] ) / (rowsum(A[r][m,:]) + eps)
// grid = 256 blocks (16-row tiles), 256 thr = 8 waves; wave w owns cols [16w,16w+16)
// A double-buffered in LDS via GLOBAL_LOAD_ASYNC_TO_LDS_B128 (ASYNCcnt-tracked);
// rowsum fused from the A fragments each lane already holds.
// ---------------------------------------------------------------------------
__global__ __launch_bounds__(256) void rgcn_agg(
    const float* __restrict__ A,    // [8][4096][4096]
    const float* __restrict__ HTf,  // packed H^T fragments
    float* __restrict__ out)        // [4096][128]
{
  __shared__ float sa[2][16 * LDA];

  const int lane = threadIdx.x & 31;
  const int wv   = threadIdx.x >> 5;
  const int m0   = blockIdx.x * 16;

  const int mrow = lane & 15;
  const int half = lane >> 4;

  // per-thread staging slots: 2 x float4 per thread covers the 16x128 chunk
  const int q0 = threadIdx.x,        row0 = q0 >> 5, cq0 = q0 & 31;
  const int q1 = threadIdx.x + 256,  row1 = q1 >> 5, cq1 = q1 & 31;

  v8f oacc = {};

  for (int r = 0; r < 8; ++r) {
    const float* Ar = A + (size_t)r * 4096 * 4096 + (size_t)m0 * 4096;
    const float* Hr = HTf + (size_t)r * 524288
                          + (size_t)(wv * 16 + mrow) * 4 + 2 * half;

    v8f   acc = {};
    float rs  = 0.0f;                    // partial rowsum (this lane's K residues)
    const int NC = 4096 / KT;            // 32 chunks

#if HAS_ASYNC_LDS
    #define STAGE_ASYNC(bufi, koff)                                            \
      do {                                                                     \
        __builtin_amdgcn_global_load_async_to_lds_b128(                        \
            (gptr_b128)(Ar + (size_t)row0 * 4096 + (koff) + 4 * cq0),          \
            (lptr_b128)&sa[bufi][row0 * LDA + 4 * cq0], 0, 0);                 \
        __builtin_amdgcn_global_load_async_to_lds_b128(                        \
            (gptr_b128)(Ar + (size_t)row1 * 4096 + (koff) + 4 * cq1),          \
            (lptr_b128)&sa[bufi][row1 * LDA + 4 * cq1], 0, 0);                 \
      } while (0)

    STAGE_ASYNC(0, 0);
    for (int c = 0; c < NC; ++c) {
      __builtin_amdgcn_s_wait_asynccnt(0);   // own chunk-c issues complete
      __syncthreads();                        // all waves' chunk-c data resident
      if (c + 1 < NC) STAGE_ASYNC((c + 1) & 1, (c + 1) * KT);  // overlap w/ compute

      const float* sbuf   = sa[c & 1];
      const int    K4base = c * (KT / 4);
      #pragma unroll
      for (int kk = 0; kk < KT / 4; ++kk) {
        v2f a = *(const v2f*)&sbuf[mrow * LDA + 4 * kk + 2 * half];
        v2f b = *(const v2f*)(Hr + (size_t)(K4base + kk) * 512);
        rs += a.x + a.y;
        acc = __builtin_amdgcn_wmma_f32_16x16x4_f32(false, a, false, b,
                                                    (short)0, acc, false, false);
      }
      __syncthreads();                        // done reading buf[c&1]
    }
    #undef STAGE_ASYNC
#else
    // fallback: register-prefetch double buffering
    {
      float4 t0 = *(const float4*)(Ar + (size_t)row0 * 4096 + 4 * cq0);
      float4 t1 = *(const float4*)(Ar + (size_t)row1 * 4096 + 4 * cq1);
      *(float4*)&sa[0][row0 * LDA + 4 * cq0] = t0;
      *(float4*)&sa[0][row1 * LDA + 4 * cq1] = t1;
    }
    __syncthreads();
    for (int c = 0; c < NC; ++c) {
      float4 p0, p1;
      if (c + 1 < NC) {
        p0 = *(const float4*)(Ar + (size_t)row0 * 4096 + (size_t)(c + 1) * KT + 4 * cq0);
        p1 = *(const float4*)(Ar + (size_t)row1 * 4096 + (size_t)(c + 1) * KT + 4 * cq1);
      }
      const float* sbuf   = sa[c & 1];
      const int    K4base = c * (KT / 4);
      #pragma unroll
      for (int kk = 0; kk < KT / 4; ++kk) {
        v2f a = *(const v2f*)&sbuf[mrow * LDA + 4 * kk + 2 * half];
        v2f b = *(const v2f*)(Hr + (size_t)(K4base + kk) * 512);
        rs += a.x + a.y;
        acc = __builtin_amdgcn_wmma_f32_16x16x4_f32(false, a, false, b,
                                                    (short)0, acc, false, false);
      }
      if (c + 1 < NC) {
        __syncthreads();
        *(float4*)&sa[(c + 1) & 1][row0 * LDA + 4 * cq0] = p0;
        *(float4*)&sa[(c + 1) & 1][row1 * LDA + 4 * cq1] = p1;
        __syncthreads();
      }
    }
    __syncthreads();
#endif

    // full rowsum for row mrow lives on lanes {mrow, mrow+16}
    const float full = rs + __shfl_xor(rs, 16, 32);
    #pragma unroll
    for (int v = 0; v < 8; ++v) {
      const float nv = __shfl(full, v + 8 * half, 32);  // norm of D-row v+8*half
      oacc[v] += acc[v] / (nv + EPS);
    }
    __syncthreads();
  }

  float* op = out + (size_t)m0 * 128 + wv * 16 + mrow;
  #pragma unroll
  for (int v = 0; v < 8; ++v) {
    op[(size_t)(v + 8 * half) * 128] = oacc[v];
  }
}

// ---------------------------------------------------------------------------
extern "C" void kernel_launch(void* const* d_in, const int* in_sizes, int n_in,
                              void* d_out, int out_size, void* d_ws, size_t ws_size,
                              hipStream_t stream) {
  const float* X = (const float*)d_in[0];   // [4096,128]
  const float* A = (const float*)d_in[1];   // [8,4096,4096]
  const float* W = (const float*)d_in[2];   // [8,128,128]
  float* out = (float*)d_out;               // [4096,128]
  float* HTf = (float*)d_ws;                // 16 MB packed H^T fragments

  rgcn_hproj<<<dim3(8 * 256), dim3(256), 0, stream>>>(X, W, HTf);
  rgcn_agg  <<<dim3(256),     dim3(256), 0, stream>>>(A, HTf, out);
}